// EngramShortConv_72739566125408
// MI455X (gfx1250) — compile-verified
//
#include <hip/hip_runtime.h>
#include <stdint.h>

// Problem constants (match reference)
constexpr int Bc  = 4;
constexpr int Lc  = 4096;
constexpr int HCc = 4;
constexpr int Dc  = 1024;
constexpr int PADc = 6;          // (K-1)*DIL
constexpr float EPSc = 1e-5f;

// Tiling
constexpr int LT   = 64;         // l-rows of output per block
constexpr int NBUF = 16;         // async pipeline depth (rows staged in LDS, 64 KB)
constexpr int NT   = 256;        // 8 waves of 32; thread owns 4 channels (vec4)

typedef float f32x4 __attribute__((ext_vector_type(4)));

// One l-row of x for this (b,hc): 1024 floats = 4KB, loaded as 256 lanes x b128,
// DMA'd straight into LDS via the gfx1250 async-to-LDS path (ASYNCcnt tracked,
// no VGPR landing zone -> deep per-wave MLP without register pressure).
__device__ __forceinline__ void issue_async_row(const float* gsrc, float* ldst) {
  unsigned long long ga = (unsigned long long)(uintptr_t)gsrc;
  // Generic LDS pointer: low 32 bits are the LDS byte address (ISA 10.2 aperture rule).
  unsigned int la = (unsigned int)(uintptr_t)ldst;
  asm volatile("global_load_async_to_lds_b128 %0, %1, off"
               :: "v"(la), "v"(ga) : "memory");
}

__global__ __launch_bounds__(NT) void engram_fused(
    const float* __restrict__ x,    // [B, L, HC, D]
    const float* __restrict__ nw,   // [HC, D]
    const float* __restrict__ cw,   // [C, 1, K], C = HC*D
    float* __restrict__ out) {      // [B, L, HC, D]
  __shared__ __align__(16) float xs[NBUF][Dc];   // 64 KB staging ring
  __shared__ float part[2][NT / 32];             // double-buffered wave partials

  const int t    = threadIdx.x;
  const int lane = t & 31;
  const int wv   = t >> 5;
  const int b    = blockIdx.x >> 2;
  const int hc   = blockIdx.x & 3;
  const int l0   = blockIdx.y * LT;
  const int d    = t << 2;                       // first of 4 owned channels

  // Per-thread constants: norm weight and 4 channels x 4 taps of conv weight.
  const f32x4 nw4 = *(const f32x4*)(nw + hc * Dc + d);
  const int c0 = hc * Dc + d;
  const f32x4 wc0 = *(const f32x4*)(cw + (size_t)(c0 + 0) * 4);  // channel-major taps
  const f32x4 wc1 = *(const f32x4*)(cw + (size_t)(c0 + 1) * 4);
  const f32x4 wc2 = *(const f32x4*)(cw + (size_t)(c0 + 2) * 4);
  const f32x4 wc3 = *(const f32x4*)(cw + (size_t)(c0 + 3) * 4);
  // Transpose to tap-major once: tK[j] = w[channel j, tap K]  -> packed vector FMAs.
  const f32x4 t0 = {wc0.x, wc1.x, wc2.x, wc3.x};
  const f32x4 t1 = {wc0.y, wc1.y, wc2.y, wc3.y};
  const f32x4 t2 = {wc0.z, wc1.z, wc2.z, wc3.z};
  const f32x4 t3 = {wc0.w, wc1.w, wc2.w, wc3.w};

  const size_t rowStride = (size_t)HCc * Dc;     // floats between consecutive l rows
  const float* xbase = x   + ((size_t)b * Lc * HCc + hc) * Dc + d;
  float*       obase = out + ((size_t)b * Lc * HCc + hc) * Dc + d;

  const int lstart = (l0 >= PADc) ? (l0 - PADc) : 0;   // always even
  const int lend   = l0 + LT - 1;

  // Prologue: fill the async pipeline to constant depth NBUF.
  for (int j = 0; j < NBUF; ++j) {
    int ls = lstart + j; if (ls > lend) ls = lend;
    issue_async_row(xbase + (size_t)ls * rowStride, &xs[j][d]);
  }

  // Dilation-2 history as two parity streams: h0 = xn[l-2], h1 = xn[l-4], h2 = xn[l-6].
  f32x4 he0 = {0.f, 0.f, 0.f, 0.f}, he1 = he0, he2 = he0;
  f32x4 ho0 = he0, ho1 = he0, ho2 = he0;

  auto step = [&](int l, int par, f32x4& h0, f32x4& h1, f32x4& h2) {
    const int bi = (l - lstart) & (NBUF - 1);
    // Oldest of the 16 in-flight rows is ours; in-order completion => <=15 outstanding.
    asm volatile("s_wait_asynccnt 0xf" ::: "memory");
    const f32x4 cur = *(const f32x4*)&xs[bi][d];
    // Recycle this buffer: our b128 LDS read must retire before the DMA overwrites it.
    {
      int ls = l + NBUF; if (ls > lend) ls = lend;   // clamped tail keeps depth constant
      asm volatile("s_wait_dscnt 0x0" ::: "memory");
      issue_async_row(xbase + (size_t)ls * rowStride, &xs[bi][d]);
    }

    // RMS reduction over D: 4 local squares -> wave32 xor-shuffle tree -> 8 partials.
    float s = cur.x * cur.x + cur.y * cur.y + cur.z * cur.z + cur.w * cur.w;
    s += __shfl_xor(s, 16, 32);
    s += __shfl_xor(s,  8, 32);
    s += __shfl_xor(s,  4, 32);
    s += __shfl_xor(s,  2, 32);
    s += __shfl_xor(s,  1, 32);
    if (lane == 0) part[par][wv] = s;
    __syncthreads();                                  // single barrier per row (dbl-buffered)
    float tot = 0.f;
#pragma unroll
    for (int i = 0; i < NT / 32; ++i) tot += part[par][i];
    const float rstd = rsqrtf(tot * (1.0f / Dc) + EPSc);

    const f32x4 xn = cur * rstd * nw4;                // packed normalize

    if (l >= l0) {
      // y[l] = w3*xn[l] + w2*xn[l-2] + w1*xn[l-4] + w0*xn[l-6], 4 packed FMAs.
      const f32x4 yv = t3 * xn + t2 * h0 + t1 * h1 + t0 * h2;
      f32x4 o4;
      o4.x = yv.x / (1.0f + __expf(-yv.x));
      o4.y = yv.y / (1.0f + __expf(-yv.y));
      o4.z = yv.z / (1.0f + __expf(-yv.z));
      o4.w = yv.w / (1.0f + __expf(-yv.w));
      // Write-once stream: non-temporal so the 256MB output doesn't evict the
      // x halo rows (re-read by adjacent l-tiles) from the 192MB L2.
      __builtin_nontemporal_store(o4, (f32x4*)(obase + (size_t)l * rowStride));
    }
    h2 = h1; h1 = h0; h0 = xn;
  };

  // lstart and the trip count are always even -> static parity, unrolled by 2.
  for (int l = lstart; l <= lend; l += 2) {
    step(l,     0, he0, he1, he2);
    step(l + 1, 1, ho0, ho1, ho2);
  }
}

extern "C" void kernel_launch(void* const* d_in, const int* /*in_sizes*/, int /*n_in*/,
                              void* d_out, int /*out_size*/, void* /*d_ws*/, size_t /*ws_size*/,
                              hipStream_t stream) {
  const float* x  = (const float*)d_in[0];
  const float* nw = (const float*)d_in[1];
  const float* cw = (const float*)d_in[2];
  float* out = (float*)d_out;
  dim3 grid(Bc * HCc, Lc / LT);   // 16 x 64 = 1024 blocks, 8192 wave32s
  dim3 block(NT);
  hipLaunchKernelGGL(engram_fused, grid, block, 0, stream, x, nw, cw, out);
}